// AtlasApan_31911607009493
// MI455X (gfx1250) — compile-verified
//
#include <hip/hip_runtime.h>
#include <math.h>

// ---- problem constants (match reference) ----
#define NUM_NODES 50000
#define DIM_EMBED 128
#define DIM_TIME  64
#define SLOTS     10
#define DIM_MSG   288            // 2*128 + 32
#define DIM_KV    352            // DIM_MSG + DIM_TIME
#define BATCH     8192
#define NTOT      (3*BATCH)      // 24576
#define TILE      16             // nodes per workgroup (WMMA M)
#define THREADS   128            // 4 waves (wave32)
#define MSG_PAD   356            // padded LDS row stride

typedef __attribute__((ext_vector_type(2))) float v2f;
typedef __attribute__((ext_vector_type(8))) float v8f;

// Async global->LDS copy of one 16-byte chunk (per lane).
// ISA 10.?: GLOBAL_LOAD_ASYNC_TO_LDS_B128, tracked by ASYNCcnt.
__device__ inline void async_ld_b128(unsigned lds_off, const void* gaddr) {
  unsigned long long ga = (unsigned long long)(size_t)gaddr;
  asm volatile("global_load_async_to_lds_b128 %0, %1, off"
               :: "v"(lds_off), "v"(ga) : "memory");
}
__device__ inline void wait_asynccnt0() {
  asm volatile("s_wait_asynccnt 0x0" ::: "memory");
}

// D(16x16) += A(16x4) * B(4x16) accumulated over ksteps of K=4.
// A from LDS (row-major, stride lda); B = W^T, i.e. b[k][j] = W[j][k],
// W row-major with leading dim ldb, streamed from global (hot in L2).
// Fragment layouts per CDNA5 ISA 7.12.2:
//   A: lane m=l&15, k0 = 2*(l>>4): holds A[m][k0], A[m][k0+1]
//   B: lane n=l&15, k0 = 2*(l>>4): holds B[k0][n], B[k0+1][n]
//   C/D: elem r -> D[r + 8*(l>>4)][l&15]
__device__ inline v8f wmma_gemm(const float* __restrict__ smA, int lda,
                                const float* __restrict__ gW, int ldb,
                                int ksteps, int jbase, int lane) {
  const int m  = lane & 15;
  const int kh = (lane >> 4) << 1;
  const int j  = jbase + m;
  v8f acc = {0.f, 0.f, 0.f, 0.f, 0.f, 0.f, 0.f, 0.f};
  for (int kb = 0; kb < ksteps; ++kb) {
    const int k0 = kb * 4 + kh;
    v2f a, b;
    a.x = smA[m * lda + k0];
    a.y = smA[m * lda + k0 + 1];
    b.x = gW[(size_t)j * ldb + k0];
    b.y = gW[(size_t)j * ldb + k0 + 1];
    acc = __builtin_amdgcn_wmma_f32_16x16x4_f32(false, a, false, b,
                                                (short)0, acc, false, false);
  }
  return acc;
}

__device__ inline void store_tile_lds(float* __restrict__ smD, int ldd,
                                      v8f acc, float bias, int lane) {
  const int n  = lane & 15;
  const int mb = (lane >> 4) << 3;
#pragma unroll
  for (int r = 0; r < 8; ++r) smD[(mb + r) * ldd + n] = acc[r] + bias;
}

__global__ __launch_bounds__(THREADS)
void tgn_update_kernel(const float* __restrict__ memory,
                       const float* __restrict__ mail,
                       const float* __restrict__ mail_ts,
                       const int*   __restrict__ mail_ptr,
                       const int*   __restrict__ mail_cnt,
                       const int*   __restrict__ nodes,
                       const float* __restrict__ curr_ts,
                       const float* __restrict__ w_q, const float* __restrict__ b_q,
                       const float* __restrict__ w_k, const float* __restrict__ b_k,
                       const float* __restrict__ w_v, const float* __restrict__ b_v,
                       const float* __restrict__ mlp_w, const float* __restrict__ mlp_b,
                       const float* __restrict__ ln_g, const float* __restrict__ ln_b,
                       const float* __restrict__ time_w, const float* __restrict__ time_b,
                       float* __restrict__ out_updated) {
  __shared__ float sm_prev[TILE][DIM_EMBED];
  __shared__ float sm_q[TILE][DIM_EMBED];
  __shared__ float sm_msg[TILE][MSG_PAD];   // msg tile; reused as LN x later
  __shared__ float sm_k[TILE][DIM_EMBED];
  __shared__ float sm_v[TILE][DIM_EMBED];
  __shared__ int   sm_node[TILE], sm_ptr[TILE], sm_cnt[TILE], sm_ord[TILE];
  __shared__ float sm_ct[TILE], sm_mts[TILE];
  __shared__ float sm_corr[2][TILE], sm_wgt[2][TILE];
  __shared__ float sm_mrun[2][TILE], sm_srun[2][TILE];
  __shared__ float sm_red[THREADS][2];
  __shared__ float sm_mu[TILE], sm_rstd[TILE];

  const int tid  = threadIdx.x;
  const int lane = tid & 31;
  const int wave = tid >> 5;
  const int nb   = blockIdx.x * TILE;

  // low 32 bits of a flat LDS pointer are the LDS byte offset
  const unsigned lds_prev = (unsigned)(size_t)&sm_prev[0][0];
  const unsigned lds_msg  = (unsigned)(size_t)&sm_msg[0][0];

  // ownership for per-node accumulation: 8 threads per node, 16 cols each
  const int own_n  = tid >> 3;
  const int own_c0 = (tid & 7) * 16;
  const int own_h  = (tid & 7) >> 2;       // cols 0..63 head0, 64..127 head1

  if (tid < TILE) {
    const int node = nodes[nb + tid];
    sm_node[tid] = node;
    sm_ptr[tid]  = mail_ptr[node];
    sm_cnt[tid]  = mail_cnt[node];
    sm_ct[tid]   = curr_ts[nb + tid];
  }
  __syncthreads();

  // stage prev_mem rows (gathered) via async memory->LDS b128
  // 16 rows x 128 f32 = 512 x 16B chunks; 4 per thread, uniform trip count
  for (int idx = tid; idx < TILE * (DIM_EMBED / 4); idx += THREADS) {
    const int n = idx >> 5, c4 = idx & 31;          // 32 chunks per row
    async_ld_b128(lds_prev + (unsigned)(n * DIM_EMBED + c4 * 4) * 4u,
                  memory + (size_t)sm_node[n] * DIM_EMBED + c4 * 4);
  }
  wait_asynccnt0();
  __syncthreads();

  // Q = prev_mem @ w_q^T + b_q   (M=16, K=128, N=128)
  for (int t = 0; t < 2; ++t) {
    const int jbase = (wave * 2 + t) * 16;
    v8f acc = wmma_gemm(&sm_prev[0][0], DIM_EMBED, w_q, DIM_EMBED,
                        DIM_EMBED / 4, jbase, lane);
    store_tile_lds(&sm_q[0][0], DIM_EMBED, acc, b_q[jbase + (lane & 15)], lane);
  }
  if (tid < 32) {
    const int h = tid >> 4, n = tid & 15;
    sm_mrun[h][n] = -INFINITY;
    sm_srun[h][n] = 0.f;
  }
  float oacc[16];
#pragma unroll
  for (int e = 0; e < 16; ++e) oacc[e] = 0.f;
  __syncthreads();

  // ---- slot loop with online softmax (no need to keep all V tiles) ----
  for (int s = 0; s < SLOTS; ++s) {
    if (tid < TILE) {
      int p = sm_ptr[tid] - s - 1;
      p %= SLOTS; if (p < 0) p += SLOTS;
      sm_ord[tid] = p;
      sm_mts[tid] = mail_ts[(size_t)sm_node[tid] * SLOTS + p];
      // prefetch next slot's mail row (global_prefetch_b8)
      if (s + 1 < SLOTS) {
        int pn = sm_ptr[tid] - s - 2;
        pn %= SLOTS; if (pn < 0) pn += SLOTS;
        __builtin_prefetch(mail + ((size_t)sm_node[tid] * SLOTS + pn) * DIM_MSG, 0, 0);
      }
    }
    __syncthreads();

    // stage gathered mail rows via async memory->LDS b128:
    // 16 rows x 288 f32 (1152B, 16B aligned) = 1152 chunks; 9 per thread
    for (int idx = tid; idx < TILE * (DIM_MSG / 4); idx += THREADS) {
      const int n = idx / (DIM_MSG / 4), c4 = idx - n * (DIM_MSG / 4);
      async_ld_b128(lds_msg + (unsigned)(n * MSG_PAD + c4 * 4) * 4u,
                    mail + ((size_t)sm_node[n] * SLOTS + sm_ord[n]) * DIM_MSG + c4 * 4);
    }
    // cos time features computed by VALU while async DMA streams the rows
    for (int idx = tid; idx < TILE * DIM_TIME; idx += THREADS) {
      const int n = idx >> 6, j = idx & 63;
      const float dt = sm_ct[n] - sm_mts[n];
      sm_msg[n][DIM_MSG + j] = cosf(dt * time_w[j] + time_b[j]);
    }
    wait_asynccnt0();
    __syncthreads();

    // K,V = msg @ w_{k,v}^T + b   (M=16, K=352, N=128)
    for (int t = 0; t < 2; ++t) {
      const int jbase = (wave * 2 + t) * 16;
      v8f ak = wmma_gemm(&sm_msg[0][0], MSG_PAD, w_k, DIM_KV, DIM_KV / 4, jbase, lane);
      store_tile_lds(&sm_k[0][0], DIM_EMBED, ak, b_k[jbase + (lane & 15)], lane);
      v8f av = wmma_gemm(&sm_msg[0][0], MSG_PAD, w_v, DIM_KV, DIM_KV / 4, jbase, lane);
      store_tile_lds(&sm_v[0][0], DIM_EMBED, av, b_v[jbase + (lane & 15)], lane);
    }
    __syncthreads();

    // per-(node,head) logit + online softmax state update
    if (tid < 32) {
      const int h = tid >> 4, n = tid & 15;
      float dot = 0.f;
      for (int d = 0; d < 64; ++d) dot += sm_q[n][h * 64 + d] * sm_k[n][h * 64 + d];
      const float l = dot * 0.125f;           // 1/sqrt(64)
      if (s < sm_cnt[n]) {
        const float m0 = sm_mrun[h][n];
        const float mn = fmaxf(m0, l);
        const float corr = __expf(m0 - mn);   // 0 on first valid slot
        const float wgt  = __expf(l - mn);
        sm_srun[h][n] = sm_srun[h][n] * corr + wgt;
        sm_mrun[h][n] = mn;
        sm_corr[h][n] = corr;
        sm_wgt[h][n]  = wgt;
      } else {
        sm_corr[h][n] = 1.f;
        sm_wgt[h][n]  = 0.f;
      }
    }
    __syncthreads();

    {
      const float corr = sm_corr[own_h][own_n];
      const float wgt  = sm_wgt[own_h][own_n];
#pragma unroll
      for (int e = 0; e < 16; ++e)
        oacc[e] = oacc[e] * corr + wgt * sm_v[own_n][own_c0 + e];
    }
    __syncthreads();   // protect sm_msg/sm_k/sm_v for next slot
  }

  // ---- finalize: normalize attention, residual, LayerNorm stats ----
  {
    const float denom = sm_srun[own_h][own_n];
    const float inv = (sm_cnt[own_n] > 0) ? (1.f / denom) : 0.f;  // count==0 -> out 0
    float sum = 0.f, sq = 0.f;
#pragma unroll
    for (int e = 0; e < 16; ++e) {
      const float x = oacc[e] * inv + sm_prev[own_n][own_c0 + e];
      oacc[e] = x;
      sum += x; sq += x * x;
    }
    sm_red[tid][0] = sum;
    sm_red[tid][1] = sq;
  }
  __syncthreads();
  if (tid < TILE) {
    float sum = 0.f, sq = 0.f;
    for (int r = 0; r < 8; ++r) { sum += sm_red[tid * 8 + r][0]; sq += sm_red[tid * 8 + r][1]; }
    const float mu  = sum * (1.f / 128.f);
    const float var = sq * (1.f / 128.f) - mu * mu;
    sm_mu[tid]   = mu;
    sm_rstd[tid] = rsqrtf(var + 1e-5f);
  }
  __syncthreads();
  {
    const float mu = sm_mu[own_n], rs = sm_rstd[own_n];
#pragma unroll
    for (int e = 0; e < 16; ++e) {
      const int c = own_c0 + e;
      sm_msg[own_n][c] = (oacc[e] - mu) * rs * ln_g[c] + ln_b[c];  // reuse sm_msg as x
    }
  }
  __syncthreads();

  // MLP: relu(x @ mlp_w^T + mlp_b) -> updated
  for (int t = 0; t < 2; ++t) {
    const int jbase = (wave * 2 + t) * 16;
    v8f acc = wmma_gemm(&sm_msg[0][0], MSG_PAD, mlp_w, DIM_EMBED,
                        DIM_EMBED / 4, jbase, lane);
    const float bias = mlp_b[jbase + (lane & 15)];
    const int n  = lane & 15;
    const int mb = (lane >> 4) << 3;
#pragma unroll
    for (int r = 0; r < 8; ++r) {
      const float v = acc[r] + bias;
      out_updated[(size_t)(nb + mb + r) * DIM_EMBED + jbase + n] = fmaxf(v, 0.f);
    }
  }
}

// ---- output assembly: copy memory/memory_ts, deterministic scatter ----
__global__ void copy_kernel(const float4* __restrict__ memory4,
                            const float* __restrict__ memory_ts,
                            float4* __restrict__ out_mem4,
                            float* __restrict__ out_ts) {
  const size_t i  = (size_t)blockIdx.x * blockDim.x + threadIdx.x;
  const size_t nm4 = (size_t)NUM_NODES * DIM_EMBED / 4;
  if (i < nm4)                      out_mem4[i]      = memory4[i];
  else if (i < nm4 + NUM_NODES)     out_ts[i - nm4]  = memory_ts[i - nm4];
}

__global__ void claim_init_kernel(int* __restrict__ claim) {
  const int i = blockIdx.x * blockDim.x + threadIdx.x;
  if (i < NUM_NODES) claim[i] = -1;
}

__global__ void claim_kernel(const int* __restrict__ nodes, int* __restrict__ claim) {
  const int i = blockIdx.x * blockDim.x + threadIdx.x;
  if (i < 2 * BATCH) atomicMax(&claim[nodes[i]], i);   // last occurrence wins
}

__global__ void scatter_kernel(const int* __restrict__ nodes,
                               const float* __restrict__ curr_ts,
                               const float4* __restrict__ updated4,
                               const int* __restrict__ claim,
                               float4* __restrict__ out_mem4,
                               float* __restrict__ out_ts) {
  const int idx = blockIdx.x * blockDim.x + threadIdx.x;
  const int i = idx >> 5, c = idx & 31;   // 32 float4 per row
  if (i < 2 * BATCH) {
    const int node = nodes[i];
    if (claim[node] == i) {
      out_mem4[(size_t)node * 32 + c] = updated4[(size_t)i * 32 + c];
      if (c == 0) out_ts[node] = curr_ts[i];
    }
  }
}

extern "C" void kernel_launch(void* const* d_in, const int* in_sizes, int n_in,
                              void* d_out, int out_size, void* d_ws, size_t ws_size,
                              hipStream_t stream) {
  const float* memory    = (const float*)d_in[0];
  const float* memory_ts = (const float*)d_in[1];
  const float* mail      = (const float*)d_in[2];
  const float* mail_ts   = (const float*)d_in[3];
  const int*   mail_ptr  = (const int*)d_in[4];
  const int*   mail_cnt  = (const int*)d_in[5];
  const int*   nodes     = (const int*)d_in[6];
  const float* curr_ts   = (const float*)d_in[7];
  const float* w_q   = (const float*)d_in[8];
  const float* b_q   = (const float*)d_in[9];
  const float* w_k   = (const float*)d_in[10];
  const float* b_k   = (const float*)d_in[11];
  const float* w_v   = (const float*)d_in[12];
  const float* b_v   = (const float*)d_in[13];
  const float* mlp_w = (const float*)d_in[14];
  const float* mlp_b = (const float*)d_in[15];
  const float* ln_g  = (const float*)d_in[16];
  const float* ln_b  = (const float*)d_in[17];
  const float* time_w = (const float*)d_in[18];
  const float* time_b = (const float*)d_in[19];

  float* out_updated = (float*)d_out;                               // [24576,128]
  float* out_mem     = out_updated + (size_t)NTOT * DIM_EMBED;      // [50000,128]
  float* out_ts      = out_mem + (size_t)NUM_NODES * DIM_EMBED;     // [50000]
  int*   claim       = (int*)d_ws;                                  // 50000 ints (200 KB)

  tgn_update_kernel<<<NTOT / TILE, THREADS, 0, stream>>>(
      memory, mail, mail_ts, mail_ptr, mail_cnt, nodes, curr_ts,
      w_q, b_q, w_k, b_k, w_v, b_v, mlp_w, mlp_b, ln_g, ln_b,
      time_w, time_b, out_updated);

  const size_t ncp = (size_t)NUM_NODES * DIM_EMBED / 4 + NUM_NODES;
  copy_kernel<<<(unsigned)((ncp + 255) / 256), 256, 0, stream>>>(
      (const float4*)memory, memory_ts, (float4*)out_mem, out_ts);
  claim_init_kernel<<<(NUM_NODES + 255) / 256, 256, 0, stream>>>(claim);
  claim_kernel<<<(2 * BATCH + 255) / 256, 256, 0, stream>>>(nodes, claim);
  scatter_kernel<<<(2 * BATCH * 32 + 255) / 256, 256, 0, stream>>>(
      nodes, curr_ts, (const float4*)out_updated, claim, (float4*)out_mem, out_ts);
}